// node_gcn_82918638616893
// MI455X (gfx1250) — compile-verified
//
#include <hip/hip_runtime.h>

typedef __attribute__((ext_vector_type(2))) float v2f;
typedef __attribute__((ext_vector_type(8))) float v8f;

#define F_IN  5
#define F_HID 64
#define F_OUT 4

// Native fire-and-forget fp32 atomic add (global_atomic_add_f32, no CAS loop).
__device__ __forceinline__ void atomic_add_f32(float* p, float v) {
  unsafeAtomicAdd(p, v);
}

// ---------------- degree / normalization ----------------

__global__ void k_deg_init(float* __restrict__ deg, int n) {
  int i = blockIdx.x * blockDim.x + threadIdx.x;
  if (i < n) deg[i] = 1.0f;  // self-loop weight 1
}

__global__ void k_deg_accum(const int* __restrict__ dst, const float* __restrict__ w,
                            float* __restrict__ deg, int e) {
  int i = blockIdx.x * blockDim.x + threadIdx.x;
  if (i < e) atomic_add_f32(&deg[dst[i]], w[i]);
}

__global__ void k_dinv(float* __restrict__ deg, int n) {
  int i = blockIdx.x * blockDim.x + threadIdx.x;
  if (i < n) deg[i] = rsqrtf(deg[i]);  // deg >= 1 always (self-loop)
}

__global__ void k_norm(const int* __restrict__ src, const int* __restrict__ dst,
                       const float* __restrict__ w, const float* __restrict__ dinv,
                       float* __restrict__ norm, int e) {
  int i = blockIdx.x * blockDim.x + threadIdx.x;
  if (i < e) norm[i] = dinv[src[i]] * w[i] * dinv[dst[i]];
}

// ---------------- GEMM1: h1[N,64] = x[N,5] @ W1^T (W1 is [64,5]) ----------------
// One wave = one 16(node) x 16(feature) tile, K=5 padded to two K=4 WMMA steps.
// All loads unconditional (always in-bounds); zero-padding via selects, no branches.

__global__ __launch_bounds__(128) void k_gemm1_wmma(const float* __restrict__ x,
                                                    const float* __restrict__ W1,
                                                    float* __restrict__ h1) {
  const int wave = threadIdx.x >> 5;   // 0..3 -> feature block
  const int lane = threadIdx.x & 31;
  const int tile = blockIdx.x;         // node tile
  const int m    = tile * 16 + (lane & 15);
  const bool lo  = lane < 16;
  const int klo  = lo ? 0 : 2;         // A/B k-pair per ISA f32 16x16x4 layout
  const int ncol = wave * 16 + (lane & 15);

  const float* xr = x  + (size_t)m * F_IN;
  const float* wr = W1 + (size_t)ncol * F_IN;  // W1^T[k][n] = W1[n][k]

  float xa = xr[klo], xb = xr[klo + 1], x4 = xr[4];
  float wa = wr[klo], wb = wr[klo + 1], w4 = wr[4];

  v2f a0; a0.x = xa; a0.y = xb;
  v2f b0; b0.x = wa; b0.y = wb;
  v2f a1; a1.x = lo ? x4 : 0.0f; a1.y = 0.0f;   // k=4 valid, k=5..7 zero
  v2f b1; b1.x = lo ? w4 : 0.0f; b1.y = 0.0f;

  v8f c = {};
  c = __builtin_amdgcn_wmma_f32_16x16x4_f32(false, a0, false, b0, (short)0, c, false, false);
  c = __builtin_amdgcn_wmma_f32_16x16x4_f32(false, a1, false, b1, (short)0, c, false, false);

  const int mbase = tile * 16 + (lo ? 0 : 8);
  const int col   = wave * 16 + (lane & 15);
#pragma unroll
  for (int v = 0; v < 8; ++v)
    h1[(size_t)(mbase + v) * F_HID + col] = c[v];
}

__global__ void k_gemm1_tail(const float* __restrict__ x, const float* __restrict__ W1,
                             float* __restrict__ h1, int m0, int n) {
  int t = blockIdx.x * blockDim.x + threadIdx.x;
  int m = m0 + t / F_HID;
  int f = t % F_HID;
  if (m >= n) return;
  float s = 0.0f;
#pragma unroll
  for (int k = 0; k < F_IN; ++k) s += x[(size_t)m * F_IN + k] * W1[(size_t)f * F_IN + k];
  h1[(size_t)m * F_HID + f] = s;
}

// ---------------- GEMM2: h2[N,4] = g1[N,64] @ W2^T (W2 is [4,64]) ----------------
// 16x16 tile per wave, only 4 N-columns valid (B zero-padded via selects),
// K-loop of 16 WMMA steps. No branches inside the loop.

__global__ __launch_bounds__(128) void k_gemm2_wmma(const float* __restrict__ g1,
                                                    const float* __restrict__ W2,
                                                    float* __restrict__ h2, int ntiles) {
  const int wave = threadIdx.x >> 5;
  const int lane = threadIdx.x & 31;
  const int tile = blockIdx.x * 4 + wave;   // wave-uniform
  if (tile >= ntiles) return;               // uniform exit: EXEC stays all-1 in live waves

  const int m    = tile * 16 + (lane & 15);
  const int klo  = (lane < 16) ? 0 : 2;
  const int ncol = lane & 15;
  const bool nv  = ncol < F_OUT;

  const float* gr = g1 + (size_t)m * F_HID;
  const float* wr = W2 + (size_t)(nv ? ncol : 0) * F_HID;  // row 0 when padded: safe

  v8f c = {};
#pragma unroll
  for (int kc = 0; kc < F_HID / 4; ++kc) {
    int k = kc * 4 + klo;
    float t0 = wr[k], t1 = wr[k + 1];            // unconditional load
    v2f a; a.x = gr[k];          a.y = gr[k + 1];
    v2f b; b.x = nv ? t0 : 0.0f; b.y = nv ? t1 : 0.0f;  // v_cndmask, no branch
    c = __builtin_amdgcn_wmma_f32_16x16x4_f32(false, a, false, b, (short)0, c, false, false);
  }

  if (nv) {
    const int mbase = tile * 16 + ((lane < 16) ? 0 : 8);
#pragma unroll
    for (int v = 0; v < 8; ++v)
      h2[(size_t)(mbase + v) * F_OUT + ncol] = c[v];
  }
}

__global__ void k_gemm2_tail(const float* __restrict__ g1, const float* __restrict__ W2,
                             float* __restrict__ h2, int m0, int n) {
  int t = blockIdx.x * blockDim.x + threadIdx.x;
  int m = m0 + t / F_OUT;
  int f = t % F_OUT;
  if (m >= n) return;
  float s = 0.0f;
#pragma unroll
  for (int k = 0; k < F_HID; ++k) s += g1[(size_t)m * F_HID + k] * W2[(size_t)f * F_HID + k];
  h2[(size_t)m * F_OUT + f] = s;
}

// ---------------- aggregation ----------------

// g[i,f4] = bias[f4] + dinv[i]^2 * h[i,f4], float4-vectorized.
// total4 = N * (F/4), F4 = F/4.
__global__ void k_affine_v4(const float4* __restrict__ h, const float* __restrict__ dinv,
                            const float4* __restrict__ bias, float4* __restrict__ g,
                            long long total4, int F4) {
  long long i = (long long)blockIdx.x * blockDim.x + threadIdx.x;
  if (i >= total4) return;
  int node = (int)(i / F4), f = (int)(i % F4);
  float di = dinv[node];
  float s = di * di;
  float4 hv = h[i];
  float4 bv = bias[f];
  float4 r;
  r.x = bv.x + s * hv.x;
  r.y = bv.y + s * hv.y;
  r.z = bv.z + s * hv.z;
  r.w = bv.w + s * hv.w;
  g[i] = r;
}

// one thread per (edge, 8-feature chunk): float4 gathers + 8 native float atomics.
// 8 consecutive threads cover one edge's full 256B row -> coalesced gather.
__global__ void k_scatter_hid(const int* __restrict__ src, const int* __restrict__ dst,
                              const float* __restrict__ norm, const float* __restrict__ h,
                              float* __restrict__ g, int e) {
  long long t = (long long)blockIdx.x * blockDim.x + threadIdx.x;
  int edge = (int)(t >> 3);
  if (edge >= e) return;
  int f8 = ((int)t & 7) * 8;
  float nm = norm[edge];
  const float* hs = h + (size_t)src[edge] * F_HID + f8;
  float*       gd = g + (size_t)dst[edge] * F_HID + f8;
  float4 v0 = *(const float4*)(hs);
  float4 v1 = *(const float4*)(hs + 4);
  atomic_add_f32(gd + 0, nm * v0.x);
  atomic_add_f32(gd + 1, nm * v0.y);
  atomic_add_f32(gd + 2, nm * v0.z);
  atomic_add_f32(gd + 3, nm * v0.w);
  atomic_add_f32(gd + 4, nm * v1.x);
  atomic_add_f32(gd + 5, nm * v1.y);
  atomic_add_f32(gd + 6, nm * v1.z);
  atomic_add_f32(gd + 7, nm * v1.w);
}

__global__ void k_relu_v4(float4* __restrict__ g, long long total4) {
  long long i = (long long)blockIdx.x * blockDim.x + threadIdx.x;
  if (i >= total4) return;
  float4 v = g[i];
  v.x = fmaxf(v.x, 0.0f);
  v.y = fmaxf(v.y, 0.0f);
  v.z = fmaxf(v.z, 0.0f);
  v.w = fmaxf(v.w, 0.0f);
  g[i] = v;
}

// one thread per edge, 4 output features
__global__ void k_scatter_out(const int* __restrict__ src, const int* __restrict__ dst,
                              const float* __restrict__ norm, const float* __restrict__ h2,
                              float* __restrict__ out, int e) {
  int i = blockIdx.x * blockDim.x + threadIdx.x;
  if (i >= e) return;
  float nm = norm[i];
  const float4 v = *(const float4*)(h2 + (size_t)src[i] * F_OUT);
  float* od = out + (size_t)dst[i] * F_OUT;
  atomic_add_f32(od + 0, nm * v.x);
  atomic_add_f32(od + 1, nm * v.y);
  atomic_add_f32(od + 2, nm * v.z);
  atomic_add_f32(od + 3, nm * v.w);
}

// ---------------- host ----------------

extern "C" void kernel_launch(void* const* d_in, const int* in_sizes, int n_in,
                              void* d_out, int out_size, void* d_ws, size_t ws_size,
                              hipStream_t stream) {
  const float* x  = (const float*)d_in[0];
  const int*   ei = (const int*)d_in[1];
  const float* ew = (const float*)d_in[2];
  const float* W1 = (const float*)d_in[5];
  const float* b1 = (const float*)d_in[6];
  const float* W2 = (const float*)d_in[7];
  const float* b2 = (const float*)d_in[8];
  float* out = (float*)d_out;

  const int N = in_sizes[0] / F_IN;
  const int E = in_sizes[2];
  const int* src = ei;
  const int* dst = ei + E;

  char* ws = (char*)d_ws;
  size_t off = 0;
  auto alloc = [&](size_t bytes) -> void* {
    off = (off + 255) & ~(size_t)255;
    void* p = ws + off;
    off += bytes;
    return p;
  };
  float* dinv = (float*)alloc((size_t)N * 4);
  float* norm = (float*)alloc((size_t)E * 4);
  float* h1   = (float*)alloc((size_t)N * F_HID * 4);
  float* g1   = (float*)alloc((size_t)N * F_HID * 4);
  float* h2   = (float*)alloc((size_t)N * F_OUT * 4);
  (void)ws_size; (void)n_in; (void)out_size;

  auto blocks = [](long long t, int b) { return dim3((unsigned)((t + b - 1) / b)); };
  dim3 b256(256);

  // normalization coefficients (computed once, used by both layers)
  k_deg_init <<<blocks(N, 256), b256, 0, stream>>>(dinv, N);
  k_deg_accum<<<blocks(E, 256), b256, 0, stream>>>(dst, ew, dinv, E);
  k_dinv     <<<blocks(N, 256), b256, 0, stream>>>(dinv, N);
  k_norm     <<<blocks(E, 256), b256, 0, stream>>>(src, dst, ew, dinv, norm, E);

  const int ntiles = N / 16, rem = N % 16;

  // layer 1
  k_gemm1_wmma<<<dim3(ntiles), dim3(128), 0, stream>>>(x, W1, h1);
  if (rem)
    k_gemm1_tail<<<blocks((long long)rem * F_HID, 256), b256, 0, stream>>>(x, W1, h1, ntiles * 16, N);
  k_affine_v4<<<blocks((long long)N * (F_HID / 4), 256), b256, 0, stream>>>(
      (const float4*)h1, dinv, (const float4*)b1, (float4*)g1,
      (long long)N * (F_HID / 4), F_HID / 4);
  k_scatter_hid<<<blocks((long long)E * 8, 256), b256, 0, stream>>>(src, dst, norm, h1, g1, E);
  k_relu_v4<<<blocks((long long)N * (F_HID / 4), 256), b256, 0, stream>>>(
      (float4*)g1, (long long)N * (F_HID / 4));

  // layer 2
  k_gemm2_wmma<<<dim3((ntiles + 3) / 4), dim3(128), 0, stream>>>(g1, W2, h2, ntiles);
  if (rem)
    k_gemm2_tail<<<blocks((long long)rem * F_OUT, 256), b256, 0, stream>>>(g1, W2, h2, ntiles * 16, N);
  k_affine_v4<<<blocks((long long)N * (F_OUT / 4), 256), b256, 0, stream>>>(
      (const float4*)h2, dinv, (const float4*)b2, (float4*)out,
      (long long)N * (F_OUT / 4), F_OUT / 4);
  k_scatter_out<<<blocks(E, 256), b256, 0, stream>>>(src, dst, norm, h2, out, E);
}